// ToPosition_31808527794318
// MI455X (gfx1250) — compile-verified
//
#include <hip/hip_runtime.h>
#include <stddef.h>

#define IMG_W   1024
#define IMG_H   1024
#define N_IMG   32
#define TW      128     // tile width  (outputs)
#define TH      16      // tile height (outputs)
#define LROWS   (TH + 2)
#define LSTRIDE 132     // floats per LDS row: 128 interior + 2 halo + 2 pad (16B-aligned rows)
#define THREADS 256
#define THRESH  0.3f

typedef float v4f __attribute__((ext_vector_type(4)));   // native vector (NT-store compatible)

#if defined(__gfx1250__) && \
    __has_builtin(__builtin_amdgcn_global_load_async_to_lds_b128) && \
    __has_builtin(__builtin_amdgcn_global_load_async_to_lds_b32)  && \
    __has_builtin(__builtin_amdgcn_s_wait_asynccnt)
#define USE_ASYNC 1
#else
#define USE_ASYNC 0
#endif

#if USE_ASYNC
typedef int v4i __attribute__((ext_vector_type(4)));
typedef __attribute__((address_space(1))) v4i* g_v4i_ptr;   // global src, b128
typedef __attribute__((address_space(3))) v4i* l_v4i_ptr;   // LDS dst,   b128
typedef __attribute__((address_space(1))) int* g_i32_ptr;   // global src, b32
typedef __attribute__((address_space(3))) int* l_i32_ptr;   // LDS dst,   b32
#endif

__device__ __forceinline__ int clampi(int v, int lo, int hi) {
    return v < lo ? lo : (v > hi ? hi : v);
}

__global__ __launch_bounds__(THREADS) void
nms3x3_kernel(const float* __restrict__ hm,
              float* __restrict__ out_kp,
              float* __restrict__ out_conf)
{
    __shared__ float tile[LROWS * LSTRIDE];

    const int b   = blockIdx.x;
    const int img = b >> 9;          // 512 tiles per image (8 x 64)
    const int rem = b & 511;
    const int ty  = rem >> 3;        // 64 tile rows
    const int tx  = rem & 7;         // 8 tile cols
    const int x0  = tx * TW;
    const int y0  = ty * TH;
    const int tid = threadIdx.x;

    const float* base = hm + (size_t)img * (IMG_W * IMG_H);

    // ---- Stage tile into LDS: 18 rows x 128 interior floats (32 x B128 per row) ----
    for (int c = tid; c < LROWS * 32; c += THREADS) {
        const int r   = c >> 5;
        const int seg = c & 31;
        const int gy  = clampi(y0 - 1 + r, 0, IMG_H - 1);
        const float* src = base + (size_t)gy * IMG_W + x0 + seg * 4;
        float*       dst = &tile[r * LSTRIDE + seg * 4];
#if USE_ASYNC
        __builtin_amdgcn_global_load_async_to_lds_b128((g_v4i_ptr)src, (l_v4i_ptr)dst, 0, 0);
#else
        *(v4f*)dst = *(const v4f*)src;
#endif
    }
    // ---- Halo columns: tile col TW = global x0-1 (clamped), col TW+1 = x0+TW (clamped) ----
    if (tid < 2 * LROWS) {
        const bool left = tid < LROWS;
        const int  r    = left ? tid : tid - LROWS;
        const int  gy   = clampi(y0 - 1 + r, 0, IMG_H - 1);
        const int  gx   = left ? clampi(x0 - 1, 0, IMG_W - 1)
                               : clampi(x0 + TW, 0, IMG_W - 1);
        const float* src = base + (size_t)gy * IMG_W + gx;
        float*       dst = &tile[r * LSTRIDE + (left ? TW : TW + 1)];
#if USE_ASYNC
        __builtin_amdgcn_global_load_async_to_lds_b32((g_i32_ptr)src, (l_i32_ptr)dst, 0, 0);
#else
        *dst = *src;
#endif
    }
#if USE_ASYNC
    __builtin_amdgcn_s_wait_asynccnt(0);   // drain this wave's async copies
#endif
    __syncthreads();                        // all waves drained before signaling

    // ---- Compute: 512 float4 groups (16 rows x 32 groups), 2 per thread ----
    for (int g = tid; g < TH * 32; g += THREADS) {
        const int row = g >> 5;
        const int c4  = (g & 31) * 4;
        const int lr  = row + 1;

        const float* r0 = &tile[(lr - 1) * LSTRIDE];
        const float* r1 = &tile[(lr    ) * LSTRIDE];
        const float* r2 = &tile[(lr + 1) * LSTRIDE];

        const v4f a0 = *(const v4f*)(r0 + c4);
        const v4f a1 = *(const v4f*)(r1 + c4);   // center row (hm values)
        const v4f a2 = *(const v4f*)(r2 + c4);

        const int li = (c4 == 0)      ? TW     : c4 - 1;
        const int ri = (c4 == TW - 4) ? TW + 1 : c4 + 4;

        const float vl = fmaxf(fmaxf(r0[li], r1[li]), r2[li]);
        const float vr = fmaxf(fmaxf(r0[ri], r1[ri]), r2[ri]);
        const float v0 = fmaxf(fmaxf(a0.x, a1.x), a2.x);
        const float v1 = fmaxf(fmaxf(a0.y, a1.y), a2.y);
        const float v2 = fmaxf(fmaxf(a0.z, a1.z), a2.z);
        const float v3 = fmaxf(fmaxf(a0.w, a1.w), a2.w);

        const float p0 = fmaxf(fmaxf(vl, v0), v1);
        const float p1 = fmaxf(fmaxf(v0, v1), v2);
        const float p2 = fmaxf(fmaxf(v1, v2), v3);
        const float p3 = fmaxf(fmaxf(v2, v3), vr);

        const bool k0 = (p0 == a1.x) && (a1.x > THRESH);
        const bool k1 = (p1 == a1.y) && (a1.y > THRESH);
        const bool k2 = (p2 == a1.z) && (a1.z > THRESH);
        const bool k3 = (p3 == a1.w) && (a1.w > THRESH);

        v4f kp, cf;
        kp.x = k0 ? 1.0f : 0.0f;  cf.x = k0 ? a1.x : 0.0f;
        kp.y = k1 ? 1.0f : 0.0f;  cf.y = k1 ? a1.y : 0.0f;
        kp.z = k2 ? 1.0f : 0.0f;  cf.z = k2 ? a1.z : 0.0f;
        kp.w = k3 ? 1.0f : 0.0f;  cf.w = k3 ? a1.w : 0.0f;

        const size_t o = (size_t)img * (IMG_W * IMG_H)
                       + (size_t)(y0 + row) * IMG_W + x0 + c4;
        // Streamed, never re-read: non-temporal stores keep L2 for the input.
        __builtin_nontemporal_store(kp, (v4f*)(out_kp   + o));
        __builtin_nontemporal_store(cf, (v4f*)(out_conf + o));
    }
}

extern "C" void kernel_launch(void* const* d_in, const int* in_sizes, int n_in,
                              void* d_out, int out_size, void* d_ws, size_t ws_size,
                              hipStream_t stream)
{
    (void)in_sizes; (void)n_in; (void)out_size; (void)d_ws; (void)ws_size;
    const float* hm  = (const float*)d_in[0];
    float*       out = (float*)d_out;
    const size_t N   = (size_t)N_IMG * IMG_W * IMG_H;   // 33,554,432

    const int tiles = N_IMG * (IMG_W / TW) * (IMG_H / TH); // 32 * 8 * 64 = 16384
    nms3x3_kernel<<<tiles, THREADS, 0, stream>>>(hm, out /*kp_map*/, out + N /*confidences*/);
}